// SimpleLSTM_33741263077481
// MI455X (gfx1250) — compile-verified
//
#include <hip/hip_runtime.h>
#include <math.h>

#define HID 9      // hidden size
#define NG  36     // 4*HID gates
#define TLEN 512   // sequence length
#define NCLS 10    // classes

typedef __attribute__((ext_vector_type(2))) float v2f;
typedef __attribute__((ext_vector_type(8))) float v8f;

// ---- fast transcendentals ---------------------------------------------------
__device__ __forceinline__ float fexp2(float x) {
#if defined(__HIP_DEVICE_COMPILE__) && __has_builtin(__builtin_amdgcn_exp2f)
  return __builtin_amdgcn_exp2f(x);
#else
  return exp2f(x);
#endif
}

__device__ __forceinline__ float frcp(float x) {
#if defined(__HIP_DEVICE_COMPILE__) && __has_builtin(__builtin_amdgcn_rcpf)
  return __builtin_amdgcn_rcpf(x);
#else
  return 1.0f / x;
#endif
}

__device__ __forceinline__ float ftanh(float x) {
#if defined(__HIP_DEVICE_COMPILE__) && __has_builtin(__builtin_amdgcn_tanhf)
  return __builtin_amdgcn_tanhf(x);
#else
  // tanh(x) = 1 - 2/(exp(2x)+1);  exp(2x) = exp2(2*log2(e)*x)
  float e = fexp2(2.8853900817779268f * x);
  return 1.0f - 2.0f * frcp(e + 1.0f);
#endif
}

__device__ __forceinline__ float fsig(float x) {
  // sigmoid(x) = 0.5 + 0.5*tanh(x/2)
  return 0.5f + 0.5f * ftanh(0.5f * x);
}

// ---- WMMA wrapper: D[16x16] += A[16x4] * B[4x16], fp32 ----------------------
__device__ __forceinline__ v8f wmma4(v2f a, v2f b, v8f c) {
  // (neg_a, A, neg_b, B, c_mod, C, reuse_a, reuse_b)
  return __builtin_amdgcn_wmma_f32_16x16x4_f32(false, a, false, b, (short)0, c,
                                               false, false);
}

// ---- operand builders -------------------------------------------------------
// A layout (16x4 fp32): lane l holds M=row (l%16); VGPR v holds K = 4c + v + 2*(l/16)
__device__ __forceinline__ void load_A(const float* __restrict__ W,
                                       v2f A[3][3], int m, int hi) {
#pragma unroll
  for (int tt = 0; tt < 3; ++tt) {
    int row = 16 * tt + m;
    bool rok = row < NG;
#pragma unroll
    for (int cc = 0; cc < 3; ++cc) {
      int col = 4 * cc + 2 * hi;
      float a0 = (rok && col     < HID) ? W[row * HID + col]     : 0.0f;
      float a1 = (rok && col + 1 < HID) ? W[row * HID + col + 1] : 0.0f;
      v2f a; a.x = a0; a.y = a1;
      A[tt][cc] = a;
    }
  }
}

// C/D layout (16x16 fp32): lane l holds N=col (l%16); VGPR r holds M = 16t + r + 8*(l/16)
__device__ __forceinline__ void load_bias(const float* __restrict__ bi,
                                          const float* __restrict__ bh,
                                          v8f bias[3], int hi) {
#pragma unroll
  for (int tt = 0; tt < 3; ++tt) {
#pragma unroll
    for (int r = 0; r < 8; ++r) {
      int row = 16 * tt + 8 * hi + r;
      bias[tt][r] = (row < NG) ? (bi[row] + bh[row]) : 0.0f;
    }
  }
}

// B layout (4x16 fp32): lane l holds N=col (l%16); VGPR v holds K = 4c + v + 2*(l/16)
__device__ __forceinline__ void build_B(const float h[HID], v2f B[3], int hi) {
#pragma unroll
  for (int cc = 0; cc < 3; ++cc) {
    float lo0 = (4 * cc     < HID) ? h[(4 * cc    ) % HID] : 0.0f;
    float lo1 = (4 * cc + 1 < HID) ? h[(4 * cc + 1) % HID] : 0.0f;
    float hi0 = (4 * cc + 2 < HID) ? h[(4 * cc + 2) % HID] : 0.0f;
    float hi1 = (4 * cc + 3 < HID) ? h[(4 * cc + 3) % HID] : 0.0f;
    v2f b;
    b.x = hi ? hi0 : lo0;
    b.y = hi ? hi1 : lo1;
    B[cc] = b;
  }
}

// ---- one LSTM step for 16 batch elements (one wave32) -----------------------
// D[48x16] = bias + W_ih*x^T + W_hh*h^T ; then elementwise i,f,g,o update.
__device__ __forceinline__ void lstm_step(const v2f Aih[3][3],
                                          const v2f Ahh[3][3],
                                          const v8f bias[3],
                                          const v2f Bx[3],
                                          float h[HID], float cst[HID],
                                          int hi) {
  v2f Bh[3];
  build_B(h, Bh, hi);

  v8f D[3];
  D[0] = bias[0]; D[1] = bias[1]; D[2] = bias[2];
#pragma unroll
  for (int tt = 0; tt < 3; ++tt) {
#pragma unroll
    for (int cc = 0; cc < 3; ++cc) {
      D[tt] = wmma4(Aih[tt][cc], Bx[cc], D[tt]);
      D[tt] = wmma4(Ahh[tt][cc], Bh[cc], D[tt]);
    }
  }

  // Own 24 gate-rows (this lane half) and partner rows via cross-half shuffle.
  float own[24], part[24];
#pragma unroll
  for (int tt = 0; tt < 3; ++tt)
#pragma unroll
    for (int r = 0; r < 8; ++r) own[tt * 8 + r] = D[tt][r];
#pragma unroll
  for (int i = 0; i < 24; ++i) part[i] = __shfl_xor(own[i], 16, 32);

  // Reassemble all 36 gate values for this lane's batch column.
  float G[NG];
#pragma unroll
  for (int row = 0; row < NG; ++row) {
    int tt = row >> 4;
    int rr = row & 15;
    int half = rr >> 3;
    int r = rr & 7;
    float a = own[tt * 8 + r];
    float b = part[tt * 8 + r];
    G[row] = (half == hi) ? a : b;
  }

  // PyTorch gate order: i, f, g, o
#pragma unroll
  for (int j = 0; j < HID; ++j) {
    float ig = fsig(G[j]);
    float fg = fsig(G[HID + j]);
    float gg = ftanh(G[2 * HID + j]);
    float og = fsig(G[3 * HID + j]);
    float cn = fg * cst[j] + ig * gg;
    cst[j] = cn;
    h[j] = og * ftanh(cn);
  }
}

// ---- fused 2-layer LSTM + FC kernel: one wave32 per 16 batch elements -------
__global__ __launch_bounds__(32) void lstm2_fc_fused(
    const float* __restrict__ x,
    const float* __restrict__ wih0, const float* __restrict__ whh0,
    const float* __restrict__ bih0, const float* __restrict__ bhh0,
    const float* __restrict__ wih1, const float* __restrict__ whh1,
    const float* __restrict__ bih1, const float* __restrict__ bhh1,
    const float* __restrict__ fcw,  const float* __restrict__ fcb,
    float* __restrict__ out) {
  const int lane = threadIdx.x;   // 0..31
  const int m  = lane & 15;       // batch column within tile
  const int hi = lane >> 4;       // lane half
  const int bglob = blockIdx.x * 16 + m;

  // Preload weights into WMMA A-layout registers (constant over all steps).
  v2f Aih0[3][3], Ahh0[3][3], Aih1[3][3], Ahh1[3][3];
  load_A(wih0, Aih0, m, hi);
  load_A(whh0, Ahh0, m, hi);
  load_A(wih1, Aih1, m, hi);
  load_A(whh1, Ahh1, m, hi);

  v8f bias0[3], bias1[3];
  load_bias(bih0, bhh0, bias0, hi);
  load_bias(bih1, bhh1, bias1, hi);

  float h0[HID], c0[HID], h1[HID], c1[HID];
#pragma unroll
  for (int j = 0; j < HID; ++j) { h0[j] = c0[j] = h1[j] = c1[j] = 0.0f; }

  const float* xrow = x + (size_t)bglob * TLEN * HID;

#pragma unroll 1
  for (int t = 0; t < TLEN; ++t) {
    const float* xp = xrow + t * HID;
    if (t + 8 < TLEN) __builtin_prefetch(xp + 8 * HID, 0, 0);

    // Layer-0 input in B layout: K = 4c + v + 2*hi  (zero-pad K>=9)
    v2f Bx[3];
#pragma unroll
    for (int cc = 0; cc < 3; ++cc) {
      int k0 = 4 * cc + 2 * hi;
      float v0 = (k0     < HID) ? xp[k0]     : 0.0f;
      float v1 = (k0 + 1 < HID) ? xp[k0 + 1] : 0.0f;
      v2f b; b.x = v0; b.y = v1;
      Bx[cc] = b;
    }

    lstm_step(Aih0, Ahh0, bias0, Bx, h0, c0, hi);

    v2f B1[3];
    build_B(h0, B1, hi);
    lstm_step(Aih1, Ahh1, bias1, B1, h1, c1, hi);
  }

  // FC on last hidden state; both halves hold identical h1, low half stores.
  if (hi == 0) {
    float* op = out + (size_t)bglob * NCLS;
#pragma unroll
    for (int k = 0; k < NCLS; ++k) {
      float acc = fcb[k];
#pragma unroll
      for (int j = 0; j < HID; ++j) acc += h1[j] * fcw[k * HID + j];
      op[k] = acc;
    }
  }
}

extern "C" void kernel_launch(void* const* d_in, const int* in_sizes, int n_in,
                              void* d_out, int out_size, void* d_ws, size_t ws_size,
                              hipStream_t stream) {
  const float* x    = (const float*)d_in[0];
  const float* wih0 = (const float*)d_in[1];
  const float* whh0 = (const float*)d_in[2];
  const float* bih0 = (const float*)d_in[3];
  const float* bhh0 = (const float*)d_in[4];
  const float* wih1 = (const float*)d_in[5];
  const float* whh1 = (const float*)d_in[6];
  const float* bih1 = (const float*)d_in[7];
  const float* bhh1 = (const float*)d_in[8];
  const float* fcw  = (const float*)d_in[9];
  const float* fcb  = (const float*)d_in[10];
  float* out = (float*)d_out;

  const int B = in_sizes[0] / (TLEN * HID);   // 4096
  dim3 grid(B / 16);
  dim3 block(32);
  lstm2_fc_fused<<<grid, block, 0, stream>>>(x, wih0, whh0, bih0, bhh0,
                                             wih1, whh1, bih1, bhh1,
                                             fcw, fcb, out);
}